// SSALayer_42348377538689
// MI455X (gfx1250) — compile-verified
//
#include <hip/hip_runtime.h>
#include <hip/hip_bf16.h>
#include <math.h>

typedef __attribute__((ext_vector_type(2))) float v2f;
typedef __attribute__((ext_vector_type(8))) float v8f;
typedef __attribute__((ext_vector_type(4))) int   v4i;

#define HW   4096
#define CIN  256
#define MID  64
#define M3   192          // f(64) + g(64) + h(64) stacked rows
#define NB   4
#define BN_EPS 1e-5f

// ---------------------------------------------------------------------------
// Async global->LDS 16-byte copy (CDNA5 GLOBAL_LOAD_ASYNC_TO_LDS_B128 path,
// tracked on ASYNCcnt).  Per-lane addresses; all call sites have full EXEC.
// Builtin signature (from probe diagnostic): (AS1 v4i*, AS3 v4i*, imm, imm).
// ---------------------------------------------------------------------------
__device__ __forceinline__ void async_copy16(const float* __restrict__ g,
                                             float* __restrict__ l) {
#if __has_builtin(__builtin_amdgcn_global_load_async_to_lds_b128)
  __builtin_amdgcn_global_load_async_to_lds_b128(
      (__attribute__((address_space(1))) v4i*)(void*)g,
      (__attribute__((address_space(3))) v4i*)(void*)l, 0, 0);
#elif __has_builtin(__builtin_amdgcn_load_to_lds)
  __builtin_amdgcn_load_to_lds((void*)g,
      (__attribute__((address_space(3))) void*)l, 16, 0, 0);
#else
  *(float4*)l = *(const float4*)g;
#endif
}

__device__ __forceinline__ void async_wait0() {
#if __has_builtin(__builtin_amdgcn_global_load_async_to_lds_b128)
#if __has_builtin(__builtin_amdgcn_s_wait_asynccnt)
  __builtin_amdgcn_s_wait_asynccnt(0);
#else
  asm volatile("s_wait_asynccnt 0x0" ::: "memory");
#endif
#endif
}

// ---------------------------------------------------------------------------
// K0: fold BatchNorm into conv weights: Wc[192,256], bc[192]
// ---------------------------------------------------------------------------
__global__ void build_wc(const float* __restrict__ Wf, const float* __restrict__ bf,
                         const float* __restrict__ gf, const float* __restrict__ bef,
                         const float* __restrict__ mf, const float* __restrict__ vf,
                         const float* __restrict__ Wg, const float* __restrict__ bg,
                         const float* __restrict__ gg, const float* __restrict__ beg,
                         const float* __restrict__ mg, const float* __restrict__ vg,
                         const float* __restrict__ Wh, const float* __restrict__ bh,
                         float* __restrict__ Wc, float* __restrict__ bc) {
  int idx = blockIdx.x * blockDim.x + threadIdx.x;   // 192*256
  if (idx >= M3 * CIN) return;
  int row = idx >> 8;
  int c   = idx & 255;
  float w, bias;
  if (row < MID) {
    float inv = gf[row] * rsqrtf(vf[row] + BN_EPS);
    w = Wf[row * CIN + c] * inv;
    bias = bf[row] * inv + bef[row] - mf[row] * inv;
  } else if (row < 2 * MID) {
    int r = row - MID;
    float inv = gg[r] * rsqrtf(vg[r] + BN_EPS);
    w = Wg[r * CIN + c] * inv;
    bias = bg[r] * inv + beg[r] - mg[r] * inv;
  } else {
    int r = row - 2 * MID;
    w = Wh[r * CIN + c];
    bias = bh[r];
  }
  Wc[idx] = w;
  if (c == 0) bc[row] = bias;
}

// ---------------------------------------------------------------------------
// WMMA f32 16x16x4 GEMM:  D[b] = A[b][M,K] @ X[b][K, 4096] (+bias, relu, +res)
// 256 threads = 8 waves; 64-col slab; 4 col-tiles x (M/16) row-tiles,
// 2 waves per col-tile.  Tiles staged via async-to-LDS.
// LDS row pads: 36 / 68 -> 16B-aligned rows, conflict-free frag reads.
// ---------------------------------------------------------------------------
template <int M, int K, bool RELU_FIRST128, bool ADDRES>
__global__ __launch_bounds__(256) void wmma_gemm(
    const float* __restrict__ Abase, long aStride, const float* __restrict__ bias,
    const float* __restrict__ Xbase, long xStride,
    const float* __restrict__ Rbase, long rStride,
    float* __restrict__ Dbase, long dStride) {
  constexpr int KC  = 32;       // K chunk held in LDS
  constexpr int MT  = M / 16;   // row tiles
  constexpr int TPW = MT / 2;   // tiles per wave (2 waves share a col tile)

  __shared__ float lA[M][KC + 4];   // stride 36
  __shared__ float lX[KC][64 + 4];  // stride 68

  const int b  = blockIdx.y;
  const int n0 = blockIdx.x * 64;
  const float* A = Abase + (long)b * aStride;
  const float* X = Xbase + (long)b * xStride;
  float*       D = Dbase + (long)b * dStride;

  const int tid     = threadIdx.x;
  const int lane    = tid & 31;
  const int wave    = tid >> 5;
  const int colTile = wave & 3;
  const int wrg     = wave >> 2;        // 0 or 1

  v8f acc[TPW] = {};

  for (int k0 = 0; k0 < K; k0 += KC) {
    __syncthreads();
    // A tile: M x 32 floats as 16B chunks (8 per row)
    for (int idx = tid; idx < M * (KC / 4); idx += 256) {
      int r = idx >> 3, c4 = (idx & 7) << 2;
      async_copy16(&A[(long)r * K + k0 + c4], &lA[r][c4]);
    }
    // X tile: 32 x 64 floats as 16B chunks (16 per row)
    for (int idx = tid; idx < KC * 16; idx += 256) {
      int r = idx >> 4, c4 = (idx & 15) << 2;
      async_copy16(&X[(long)(k0 + r) * HW + n0 + c4], &lX[r][c4]);
    }
    async_wait0();
    __syncthreads();

#pragma unroll
    for (int kk = 0; kk < KC; kk += 4) {
      const int kb  = kk + (lane >> 4) * 2;        // A/B 16x4 frag K layout
      const int col = colTile * 16 + (lane & 15);
      v2f bfrag;
      bfrag.x = lX[kb][col];
      bfrag.y = lX[kb + 1][col];
#pragma unroll
      for (int i = 0; i < TPW; ++i) {
        const int row = (i * 2 + wrg) * 16 + (lane & 15);
        v2f afrag;
        afrag.x = lA[row][kb];
        afrag.y = lA[row][kb + 1];
        acc[i] = __builtin_amdgcn_wmma_f32_16x16x4_f32(
            false, afrag, false, bfrag, (short)0, acc[i], false, false);
      }
    }
  }

  // Epilogue.  C/D layout: VGPR j -> row j (lanes 0-15) / row j+8 (lanes 16-31)
  const int col   = n0 + colTile * 16 + (lane & 15);
  const int rbase = (lane >> 4) * 8;
#pragma unroll
  for (int i = 0; i < TPW; ++i) {
    const int rowTile = (i * 2 + wrg) * 16;
#pragma unroll
    for (int j = 0; j < 8; ++j) {
      const int row = rowTile + rbase + j;
      float v = acc[i][j] + bias[row];
      if (RELU_FIRST128 && row < 128) v = fmaxf(v, 0.0f);
      if (ADDRES) v += Rbase[(long)b * rStride + (long)row * HW + col];
      D[(long)row * HW + col] = v;
    }
  }
}

// ---------------------------------------------------------------------------
// K3: M[b][c1,c2] = sum_m g[b,c1,m] * h[b,c2,m]   (64x64, K=4096), WMMA.
// One block (8 waves) per batch; 16 output tiles, 2 per wave.
// ---------------------------------------------------------------------------
__global__ __launch_bounds__(256) void wmma_ght(const float* __restrict__ fgh,
                                                float* __restrict__ Mout) {
  __shared__ float lG[64][36];
  __shared__ float lH[64][36];
  const int b = blockIdx.x;
  const float* g = fgh + (long)b * M3 * HW + (long)MID * HW;
  const float* h = fgh + (long)b * M3 * HW + (long)(2 * MID) * HW;
  const int tid = threadIdx.x;
  const int lane = tid & 31;
  const int wave = tid >> 5;

  v8f acc[2] = {};

  for (int k0 = 0; k0 < HW; k0 += 32) {
    __syncthreads();
    for (int idx = tid; idx < 64 * 8; idx += 256) {   // 16B chunks, 8 per row
      int r = idx >> 3, c4 = (idx & 7) << 2;
      async_copy16(&g[(long)r * HW + k0 + c4], &lG[r][c4]);
      async_copy16(&h[(long)r * HW + k0 + c4], &lH[r][c4]);
    }
    async_wait0();
    __syncthreads();
#pragma unroll
    for (int kk = 0; kk < 32; kk += 4) {
      const int kb = kk + (lane >> 4) * 2;
#pragma unroll
      for (int t = 0; t < 2; ++t) {
        const int tile = wave + t * 8;   // 0..15
        const int mr = tile >> 2, nc = tile & 3;
        v2f afrag, bfrag;
        afrag.x = lG[mr * 16 + (lane & 15)][kb];
        afrag.y = lG[mr * 16 + (lane & 15)][kb + 1];
        bfrag.x = lH[nc * 16 + (lane & 15)][kb];    // B = h^T: B[k,n] = h[n,k]
        bfrag.y = lH[nc * 16 + (lane & 15)][kb + 1];
        acc[t] = __builtin_amdgcn_wmma_f32_16x16x4_f32(
            false, afrag, false, bfrag, (short)0, acc[t], false, false);
      }
    }
  }

  float* Mb = Mout + b * 64 * 64;
  const int rbase = (lane >> 4) * 8;
#pragma unroll
  for (int t = 0; t < 2; ++t) {
    const int tile = wave + t * 8;
    const int mr = tile >> 2, nc = tile & 3;
    const int colOut = nc * 16 + (lane & 15);
#pragma unroll
    for (int j = 0; j < 8; ++j) {
      Mb[(mr * 16 + rbase + j) * 64 + colOut] = acc[t][j];
    }
  }
}

// ---------------------------------------------------------------------------
// Small reduction kernels (power-iteration algebra; O(10 MFLOP) total)
// ---------------------------------------------------------------------------
__device__ __forceinline__ float block_reduce256(float s, float* red, int tid) {
  red[tid] = s;
  __syncthreads();
  for (int off = 128; off > 0; off >>= 1) {
    if (tid < off) red[tid] += red[tid + off];
    __syncthreads();
  }
  return red[0];
}

// t[b][c] = sum_m g[b,c,m] * v0[b,m]
__global__ void gv0_kernel(const float* __restrict__ fgh, const float* __restrict__ v0,
                           float* __restrict__ tvec) {
  __shared__ float red[256];
  int b = blockIdx.y, c = blockIdx.x;
  const float* g = fgh + (long)b * M3 * HW + (long)(MID + c) * HW;
  const float* v = v0 + (long)b * HW;
  float s = 0.f;
  for (int m = threadIdx.x; m < HW; m += 256) s += g[m] * v[m];
  s = block_reduce256(s, red, threadIdx.x);
  if (threadIdx.x == 0) tvec[b * MID + c] = s;
}

// vraw[b][n] = sum_c f[b,c,n] * t[b][c]
__global__ void vraw_kernel(const float* __restrict__ fgh, const float* __restrict__ tvec,
                            float* __restrict__ vraw) {
  __shared__ float lt[MID];
  int b = blockIdx.y;
  int n = blockIdx.x * 256 + threadIdx.x;
  if (threadIdx.x < MID) lt[threadIdx.x] = tvec[b * MID + threadIdx.x];
  __syncthreads();
  const float* f = fgh + (long)b * M3 * HW;
  float s = 0.f;
#pragma unroll
  for (int c = 0; c < MID; ++c) s += f[(long)c * HW + n] * lt[c];
  vraw[(long)b * HW + n] = s;
}

// normsq[b] = sum_n vraw[b][n]^2   (no atomics -> deterministic)
__global__ void normsq_kernel(const float* __restrict__ vraw, float* __restrict__ normsq) {
  __shared__ float red[256];
  int b = blockIdx.x;
  const float* v = vraw + (long)b * HW;
  float s = 0.f;
  for (int m = threadIdx.x; m < HW; m += 256) { float t = v[m]; s += t * t; }
  s = block_reduce256(s, red, threadIdx.x);
  if (threadIdx.x == 0) normsq[b] = s;
}

// fvgv[b][r] = sum_n fgh[b,r,n] * vraw[b,n]  for r in [0,128) (= f rows then g rows)
__global__ void fvgv_kernel(const float* __restrict__ fgh, const float* __restrict__ vraw,
                            float* __restrict__ fvgv) {
  __shared__ float red[256];
  int b = blockIdx.y, r = blockIdx.x;
  const float* rowp = fgh + (long)b * M3 * HW + (long)r * HW;
  const float* v = vraw + (long)b * HW;
  float s = 0.f;
  for (int m = threadIdx.x; m < HW; m += 256) s += rowp[m] * v[m];
  s = block_reduce256(s, red, threadIdx.x);
  if (threadIdx.x == 0) fvgv[b * 128 + r] = s;
}

// sinv[b] = normsq[b] / (fv . gv)    ( = 1/s with v normalized )
__global__ void s_kernel(const float* __restrict__ fvgv, const float* __restrict__ normsq,
                         float* __restrict__ sinv) {
  __shared__ float red[64];
  int b = blockIdx.x, t = threadIdx.x;
  red[t] = fvgv[b * 128 + t] * fvgv[b * 128 + MID + t];
  __syncthreads();
  for (int off = 32; off > 0; off >>= 1) {
    if (t < off) red[t] += red[t + off];
    __syncthreads();
  }
  if (t == 0) sinv[b] = normsq[b] / red[0];
}

// Q[b][o,c1] = sinv[b] * sum_c2 Wv[o,c2] * M[b][c1,c2]
__global__ void q_kernel(const float* __restrict__ Wv, const float* __restrict__ Mmat,
                         const float* __restrict__ sinv, float* __restrict__ Q) {
  __shared__ float lM[64 * 64];
  int b = blockIdx.y;
  for (int i = threadIdx.x; i < 64 * 64; i += 256) lM[i] = Mmat[b * 64 * 64 + i];
  __syncthreads();
  int idx = blockIdx.x * 256 + threadIdx.x;   // 256*64 outputs
  int o = idx >> 6, c1 = idx & 63;
  float inv = sinv[b];
  float s = 0.f;
#pragma unroll
  for (int c2 = 0; c2 < MID; ++c2) s += Wv[o * MID + c2] * lM[c1 * MID + c2];
  Q[((long)b * CIN + o) * MID + c1] = s * inv;
}

// ---------------------------------------------------------------------------
extern "C" void kernel_launch(void* const* d_in, const int* in_sizes, int n_in,
                              void* d_out, int out_size, void* d_ws, size_t ws_size,
                              hipStream_t stream) {
  (void)in_sizes; (void)n_in; (void)out_size; (void)ws_size;
  const float* x   = (const float*)d_in[0];
  const float* Wf  = (const float*)d_in[1];
  const float* bf  = (const float*)d_in[2];
  const float* gf  = (const float*)d_in[3];
  const float* bef = (const float*)d_in[4];
  const float* mf  = (const float*)d_in[5];
  const float* vf  = (const float*)d_in[6];
  const float* Wg  = (const float*)d_in[7];
  const float* bg  = (const float*)d_in[8];
  const float* gg  = (const float*)d_in[9];
  const float* beg = (const float*)d_in[10];
  const float* mg  = (const float*)d_in[11];
  const float* vg  = (const float*)d_in[12];
  const float* Wh  = (const float*)d_in[13];
  const float* bh  = (const float*)d_in[14];
  const float* Wv  = (const float*)d_in[15];
  const float* bv  = (const float*)d_in[16];
  const float* v0  = (const float*)d_in[17];
  float* out = (float*)d_out;

  float* ws     = (float*)d_ws;
  float* fgh    = ws;                                  // B*192*HW
  float* tvec   = fgh    + (size_t)NB * M3 * HW;       // B*64
  float* vraw   = tvec   + (size_t)NB * MID;           // B*HW
  float* normsq = vraw   + (size_t)NB * HW;            // B
  float* fvgv   = normsq + (size_t)NB;                 // B*128
  float* sinv   = fvgv   + (size_t)NB * 128;           // B
  float* Mmat   = sinv   + (size_t)NB;                 // B*64*64
  float* Q      = Mmat   + (size_t)NB * 64 * 64;       // B*256*64 (16B aligned)
  float* Wc     = Q      + (size_t)NB * CIN * MID;     // 192*256  (16B aligned)
  float* bc     = Wc     + (size_t)M3 * CIN;           // 192

  // 0) fold BN into conv weights
  build_wc<<<dim3((M3 * CIN) / 256), 256, 0, stream>>>(
      Wf, bf, gf, bef, mf, vf, Wg, bg, gg, beg, mg, vg, Wh, bh, Wc, bc);

  // 1) fgh[b] = relu128(Wc @ x[b] + bc)     [192 x 4096] per batch, K=256
  wmma_gemm<M3, CIN, true, false><<<dim3(HW / 64, NB), 256, 0, stream>>>(
      Wc, 0L, bc, x, (long)CIN * HW, nullptr, 0L, fgh, (long)M3 * HW);

  // 2) low-rank power-iteration algebra (z never materialized)
  gv0_kernel   <<<dim3(MID, NB),      256, 0, stream>>>(fgh, v0, tvec);
  vraw_kernel  <<<dim3(HW / 256, NB), 256, 0, stream>>>(fgh, tvec, vraw);
  normsq_kernel<<<NB,                 256, 0, stream>>>(vraw, normsq);
  fvgv_kernel  <<<dim3(128, NB),      256, 0, stream>>>(fgh, vraw, fvgv);
  s_kernel     <<<NB,                 64,  0, stream>>>(fvgv, normsq, sinv);

  // 3) M[b] = g h^T  (64x64, K=4096)
  wmma_ght<<<NB, 256, 0, stream>>>(fgh, Mmat);

  // 4) Q[b] = (1/s) * Wv @ M[b]^T   [256 x 64]
  q_kernel<<<dim3(64, NB), 256, 0, stream>>>(Wv, Mmat, sinv, Q);

  // 5) out[b] = Q[b] @ f[b] + bv + x[b]    [256 x 4096] per batch, K=64
  wmma_gemm<CIN, MID, false, true><<<dim3(HW / 64, NB), 256, 0, stream>>>(
      Q, (long)CIN * MID, bv, fgh, (long)M3 * HW, x, (long)CIN * HW,
      out, (long)CIN * HW);
}